// Graph_18854906429791
// MI455X (gfx1250) — compile-verified
//
#include <hip/hip_runtime.h>
#include <math.h>

#define N_INPUTS 2048
#define UNITS    2048
#define LAYERS   8
#define FANIN    4096
#define TOTAL    (N_INPUTS + LAYERS * UNITS)   // 18432

#define KSPLIT   4                  // K-dimension split across blockIdx.y
#define KCHUNK   (FANIN / KSPLIT)   // 1024 k-rows per block
#define TPB      256                // 8 waves (wave32)
#define WAVES    (TPB / 32)
#define KWAVE    (KCHUNK / WAVES)   // 128 k-rows per wave
#define GPT      (KCHUNK / TPB)     // 4 gathers per thread

typedef float v2f __attribute__((ext_vector_type(2)));
typedef float v8f __attribute__((ext_vector_type(8)));

// outputs[i] = x[i] for i<2048 else 0  (state vector must start zeroed each call)
__global__ void init_outputs(const float* __restrict__ x, float* __restrict__ outputs) {
    int i = blockIdx.x * TPB + threadIdx.x;
    if (i < TOTAL) outputs[i] = (i < N_INPUTS) ? x[i] : 0.0f;
}

// One (u-tile=16, k-chunk=1024) partial GEMV per block using V_WMMA_F32_16X16X4_F32.
// grid = (UNITS/16, KSPLIT), block = 256 threads = 8 waves, each wave covers 128 k-rows.
__global__ __launch_bounds__(TPB) void layer_gemv_wmma(
    const float* __restrict__ W,        // [FANIN][UNITS] for this layer
    const int*   __restrict__ inds,     // [FANIN] for this layer
    const float* __restrict__ outputs,  // [TOTAL] shared state
    float*       __restrict__ partial)  // [KSPLIT][UNITS] partial sums
{
    __shared__ float g[KCHUNK];         // gathered slice of the state vector (4 KB)
    __shared__ float red[WAVES][16];    // cross-wave reduction

    const int tid   = threadIdx.x;
    const int u0    = blockIdx.x * 16;
    const int kbase = blockIdx.y * KCHUNK;

    // Gather, two-phase so 4 index loads then 4 gathers are each in flight
    // together (avoids one s_wait_loadcnt 0 per element). All L2-resident.
    int idx[GPT];
    #pragma unroll
    for (int j = 0; j < GPT; ++j) idx[j] = inds[kbase + tid + j * TPB];
    float gv[GPT];
    #pragma unroll
    for (int j = 0; j < GPT; ++j) gv[j] = outputs[idx[j]];
    #pragma unroll
    for (int j = 0; j < GPT; ++j) g[tid + j * TPB] = gv[j];
    __syncthreads();

    const int wave  = tid >> 5;
    const int lane  = tid & 31;
    const int n     = lane & 15;          // N index (output unit within tile)
    const int khalf = (lane >> 4) << 1;   // lanes 16-31 carry K+2/K+3 per ISA layout

    // Dual accumulators break the WMMA C->C dependency chain.
    v8f acc0 = {};
    v8f acc1 = {};
    const int klocal0 = wave * KWAVE;
    const float* wrow = W + (size_t)(kbase + klocal0 + khalf) * UNITS + u0 + n;

    #pragma unroll 4
    for (int k0 = 0; k0 < KWAVE; k0 += 8) {
        // B tiles: 2x (4x16) of W; per-lane dword loads, 64B-coalesced per half-wave.
        v2f B0, B1;
        B0.x = wrow[0];
        B0.y = wrow[UNITS];
        B1.x = wrow[4 * UNITS];
        B1.y = wrow[5 * UNITS];
        // Prefetch 2 steps (16 rows = 128 KB of stream) ahead.
        __builtin_prefetch(wrow + 16 * UNITS, 0, 0);

        // A tiles: row M=0 holds g[k..k+3] (lane 0 -> K0,K1; lane 16 -> K2,K3),
        // rows 1..15 zero. 16x compute waste is irrelevant: HBM-bound by ~8x.
        const int kl = klocal0 + k0 + khalf;
        v2f A0, A1;
        A0.x = (n == 0) ? g[kl]     : 0.0f;
        A0.y = (n == 0) ? g[kl + 1] : 0.0f;
        A1.x = (n == 0) ? g[kl + 4] : 0.0f;
        A1.y = (n == 0) ? g[kl + 5] : 0.0f;

        acc0 = __builtin_amdgcn_wmma_f32_16x16x4_f32(
            false, A0, false, B0, (short)0, acc0, false, false);
        acc1 = __builtin_amdgcn_wmma_f32_16x16x4_f32(
            false, A1, false, B1, (short)0, acc1, false, false);

        wrow += 8 * UNITS;
    }

    // Row M=0 of C lives in VGPR0 (acc[0]) on lanes 0..15, N = lane.
    if (lane < 16) red[wave][lane] = acc0[0] + acc1[0];
    __syncthreads();

    if (tid < 16) {
        float s = 0.0f;
        #pragma unroll
        for (int w = 0; w < WAVES; ++w) s += red[w][tid];   // fixed order: deterministic
        partial[(size_t)blockIdx.y * UNITS + u0 + tid] = s;
    }
}

// y[u] = tanh(b[u] + sum_s partial[s][u]); fixed order -> deterministic.
__global__ void layer_finish(const float* __restrict__ partial,
                             const float* __restrict__ bias,
                             float*       __restrict__ ydst) {
    int u = blockIdx.x * TPB + threadIdx.x;
    if (u < UNITS) {
        float s = bias[u];
        #pragma unroll
        for (int sdx = 0; sdx < KSPLIT; ++sdx)
            s += partial[(size_t)sdx * UNITS + u];
        ydst[u] = tanhf(s);
    }
}

extern "C" void kernel_launch(void* const* d_in, const int* in_sizes, int n_in,
                              void* d_out, int out_size, void* d_ws, size_t ws_size,
                              hipStream_t stream) {
    const float* x         = (const float*)d_in[0];   // [2048]
    const int*   node_inds = (const int*)  d_in[1];   // [8][4096]
    const float* Ws        = (const float*)d_in[2];   // [8][4096][2048]
    const float* bs        = (const float*)d_in[3];   // [8][2048]
    float* out = (float*)d_out;                       // [2048]

    float* outputs = (float*)d_ws;                    // TOTAL floats (72 KB)
    float* partial = outputs + TOTAL;                 // KSPLIT*UNITS floats (32 KB)

    init_outputs<<<(TOTAL + TPB - 1) / TPB, TPB, 0, stream>>>(x, outputs);

    for (int i = 0; i < LAYERS; ++i) {
        const float* W    = Ws + (size_t)i * FANIN * UNITS;
        const float* bias = bs + (size_t)i * UNITS;
        const int*   inds = node_inds + (size_t)i * FANIN;
        // Layers 0..6 append into the state vector; layer 7 writes the final output.
        float* ydst = (i == LAYERS - 1) ? out
                                        : (outputs + (size_t)(i + 1) * N_INPUTS);

        dim3 grid(UNITS / 16, KSPLIT);   // 128 u-tiles x 4 k-slices = 512 blocks
        layer_gemv_wmma<<<grid, TPB, 0, stream>>>(W, inds, outputs, partial);
        layer_finish<<<(UNITS + TPB - 1) / TPB, TPB, 0, stream>>>(partial, bias, ydst);
    }
}